// GIN_21784074125533
// MI455X (gfx1250) — compile-verified
//
#include <hip/hip_runtime.h>

#define N_NODES 100000
#define F 128
#define BN_EPS 1e-5f

typedef __attribute__((ext_vector_type(2))) float v2f;
typedef __attribute__((ext_vector_type(8))) float v8f;

// ---------------------------------------------------------------- utilities

__global__ void zero_kernel(float* __restrict__ p, int n4) {
    int i = blockIdx.x * blockDim.x + threadIdx.x;
    if (i < n4) {
        float4 z; z.x = 0.f; z.y = 0.f; z.z = 0.f; z.w = 0.f;
        ((float4*)p)[i] = z;
    }
}

// One wave (32 lanes) per edge; each lane moves 4 consecutive features.
// agg[dst[e]] += feat[src[e]]
__global__ void scatter_kernel(const float* __restrict__ feat,
                               const int* __restrict__ src,
                               const int* __restrict__ dst,
                               float* __restrict__ agg, int nEdges) {
    int t = blockIdx.x * blockDim.x + threadIdx.x;
    int e = t >> 5;
    int lane = t & 31;
    if (e >= nEdges) return;
    int s = src[e];
    int d = dst[e];
    float4 v = ((const float4*)(feat + (size_t)s * F))[lane];
    float* o = agg + (size_t)d * F + lane * 4;
    atomicAdd(o + 0, v.x);
    atomicAdd(o + 1, v.y);
    atomicAdd(o + 2, v.z);
    atomicAdd(o + 3, v.w);
}

// ------------------------------------------------------------- WMMA GEMM
// out[m, n] = epilogue( sum_k (Agg[m,k] + Xb[m,k]) * W[k,n] + bias[n] )
// Block: 256 threads = 8 waves, 128 rows per block, 16 rows per wave.
//
// LDS layout:
//  sW: pair-interleaved weights. Pair-row p holds {W[2p][n], W[2p+1][n]}
//      at float index p*288 + 2n (stride 288 floats => lane-half bank sets
//      disjoint; each B fragment is one aligned ds_load_b64, and adjacent
//      j-tiles merge into ds_load_2addr_b64).
//  sX: X tile, row stride 132 floats (banks (4m + k) % 64 => conflict-free
//      ds_load_b64 A fragments). Agg is async-staged straight into LDS via
//      global_load_async_to_lds_b128; base features Xb are added in place.
#define SW_STRIDE 288              // floats per pair-row (256 used + 32 pad)
#define SX_STRIDE 132              // floats per row (128 used + 4 pad)

__global__ void __launch_bounds__(256)
gemm_kernel(const float* __restrict__ Agg, const float* __restrict__ Xb,
            const float* __restrict__ W, const float* __restrict__ bias,
            const float* __restrict__ gamma, const float* __restrict__ beta,
            const float* __restrict__ rmean, const float* __restrict__ rvar,
            float* __restrict__ out, int nRows, int doBN) {
    __shared__ float sW[64 * SW_STRIDE];     // 73,728 B
    __shared__ float sX[128 * SX_STRIDE];    // 67,584 B

    const int tid = threadIdx.x;
    const int block0 = blockIdx.x * 128;

    // ---- async-stage Agg tile into sX (16B chunks, padded rows) ----
    const unsigned sxBase = (unsigned)(size_t)(&sX[0]);   // low 32b = LDS offset
    #pragma unroll
    for (int t = 0; t < 16; ++t) {
        int i = tid + 256 * t;
        int r = i >> 5;            // tile row 0..127
        int c = i & 31;            // 16B chunk 0..31
        int grow = block0 + r; if (grow >= nRows) grow = nRows - 1;
        unsigned ldsAddr = sxBase + (unsigned)(r * SX_STRIDE + c * 4) * 4u;
        unsigned long long ga =
            (unsigned long long)(size_t)(Agg + (size_t)grow * F + c * 4);
        asm volatile("global_load_async_to_lds_b128 %0, %1, off"
                     :: "v"(ldsAddr), "v"(ga) : "memory");
    }

    // ---- stage W pair-interleaved (overlaps with async above) ----
    for (int i = tid; i < 64 * F; i += 256) {
        int p = i >> 7;            // pair-row 0..63
        int n = i & 127;           // column
        float w0v = W[(size_t)(2 * p) * F + n];
        float w1v = W[(size_t)(2 * p + 1) * F + n];
        sW[p * SW_STRIDE + 2 * n]     = w0v;
        sW[p * SW_STRIDE + 2 * n + 1] = w1v;
    }

    // ---- wait own async chunks, then fuse-add base features in LDS ----
    asm volatile("s_wait_asynccnt 0x0" ::: "memory");
    #pragma unroll
    for (int t = 0; t < 16; ++t) {
        int i = tid + 256 * t;
        int r = i >> 5;
        int c = i & 31;
        int grow = block0 + r; if (grow >= nRows) grow = nRows - 1;
        float4 xv = *(const float4*)(Xb + (size_t)grow * F + c * 4);
        float4* sp = (float4*)&sX[r * SX_STRIDE + c * 4];
        float4 cur = *sp;
        cur.x += xv.x; cur.y += xv.y; cur.z += xv.z; cur.w += xv.w;
        *sp = cur;
    }
    __syncthreads();

    // ---- per-wave 16x128 WMMA strip ----
    const int wv  = tid >> 5;
    const int lane = tid & 31;
    const int hf  = lane >> 4;
    const int l16 = lane & 15;
    const int rowBase = block0 + wv * 16;

    // Incrementing pointers: strength-reduce per-iteration address math.
    const float* aPtr = &sX[(wv * 16 + l16) * SX_STRIDE + 2 * hf];
    const float* bPtr = &sW[hf * SW_STRIDE + 2 * l16];

    v8f acc[8] = {};
    for (int k = 0; k < F; k += 4) {
        // A fragment (16x4): VGPR0 = K=k / K=k+2, VGPR1 = K=k+1 / K=k+3
        v2f a = *(const v2f*)aPtr;
        #pragma unroll
        for (int j = 0; j < 8; ++j) {
            v2f b = *(const v2f*)(bPtr + j * 32);   // ds_load_b64 (pairs merge)
            acc[j] = __builtin_amdgcn_wmma_f32_16x16x4_f32(
                false, a, false, b, (short)0, acc[j], false, false);
        }
        aPtr += 4;                  // next K group
        bPtr += 2 * SW_STRIDE;      // next pair-row group
    }

    // ---- epilogue: bias (+ BatchNorm eval + ReLU), masked store ----
    #pragma unroll
    for (int j = 0; j < 8; ++j) {
        int col = j * 16 + l16;
        float sc = 1.f;
        float sh = bias[col];
        if (doBN) {
            float s = gamma[col] * rsqrtf(rvar[col] + BN_EPS);
            sh = (bias[col] - rmean[col]) * s + beta[col];
            sc = s;
        }
        #pragma unroll
        for (int r = 0; r < 8; ++r) {
            // C/D layout: VGPR r -> M = r (lanes 0-15) / r+8 (lanes 16-31)
            int row = rowBase + r + 8 * hf;
            if (row < nRows) {
                float v = acc[j][r] * sc + sh;
                if (doBN) v = v > 0.f ? v : 0.f;
                out[(size_t)row * F + col] = v;
            }
        }
    }
}

// ---------------------------------------------------------------- launcher

extern "C" void kernel_launch(void* const* d_in, const int* in_sizes, int n_in,
                              void* d_out, int out_size, void* d_ws, size_t ws_size,
                              hipStream_t stream) {
    const float* x     = (const float*)d_in[0];
    const int*   ei    = (const int*)d_in[1];
    const float* w0    = (const float*)d_in[2];
    const float* b0    = (const float*)d_in[3];
    const float* gamma = (const float*)d_in[4];
    const float* beta  = (const float*)d_in[5];
    const float* rmean = (const float*)d_in[6];
    const float* rvar  = (const float*)d_in[7];
    const float* w1    = (const float*)d_in[8];
    const float* b1    = (const float*)d_in[9];
    float* out = (float*)d_out;

    const int E = in_sizes[1] / 2;
    const int* src = ei;          // edge_index[0, :]
    const int* dst = ei + E;      // edge_index[1, :]

    float* agg = (float*)d_ws;                       // N*128 floats
    float* h   = agg + (size_t)N_NODES * F;          // N*128 floats

    const int n4 = (N_NODES * F) / 4;
    const int zb = (n4 + 255) / 256;
    const int sb = (E * 32 + 255) / 256;             // wave-per-edge
    const int gb = (N_NODES + 127) / 128;

    // Layer 0: agg = segsum(x); h = relu(bn((x + agg) @ w0 + b0))
    zero_kernel<<<zb, 256, 0, stream>>>(agg, n4);
    scatter_kernel<<<sb, 256, 0, stream>>>(x, src, dst, agg, E);
    gemm_kernel<<<gb, 256, 0, stream>>>(agg, x, w0, b0, gamma, beta, rmean, rvar,
                                        h, N_NODES, 1);

    // Layer 1: agg = segsum(h); out = (h + agg) @ w1 + b1
    zero_kernel<<<zb, 256, 0, stream>>>(agg, n4);
    scatter_kernel<<<sb, 256, 0, stream>>>(h, src, dst, agg, E);
    gemm_kernel<<<gb, 256, 0, stream>>>(agg, h, w1, b1, nullptr, nullptr, nullptr, nullptr,
                                        out, N_NODES, 0);
}